// Triplane_1906965479801
// MI455X (gfx1250) — compile-verified
//
#include <hip/hip_runtime.h>
#include <stdint.h>

#define RES 512
#define CH  32
#define PLANE_ELEMS (RES * RES * CH)       // 8,388,608 elements per plane (channel-last)
#define TRIPLANE_ELEMS (3 * PLANE_ELEMS)   // 25,165,824 elements

// ---------------------------------------------------------------------------
// Bilinear sample from channel-last layout wsT[((p*RES + y)*RES + x)*CH + c]
// gx maps to W (x), gy maps to H (y); align_corners=True; zero padding.
// ---------------------------------------------------------------------------
__device__ __forceinline__ float sample_plane_cl(const float* __restrict__ t,
                                                 uint32_t planeBase,
                                                 float gx, float gy, int lane) {
  float x = (gx + 1.0f) * (0.5f * (float)(RES - 1));
  float y = (gy + 1.0f) * (0.5f * (float)(RES - 1));
  float xf = floorf(x), yf = floorf(y);
  float wx1 = x - xf, wy1 = y - yf;
  float wx0 = 1.0f - wx1, wy0 = 1.0f - wy1;
  int x0 = (int)xf, y0 = (int)yf;
  int x1 = x0 + 1, y1 = y0 + 1;
  float ax0 = (x0 >= 0 && x0 < RES) ? wx0 : 0.0f;
  float ax1 = (x1 >= 0 && x1 < RES) ? wx1 : 0.0f;
  float ay0 = (y0 >= 0 && y0 < RES) ? wy0 : 0.0f;
  float ay1 = (y1 >= 0 && y1 < RES) ? wy1 : 0.0f;
  int x0c = min(max(x0, 0), RES - 1), x1c = min(max(x1, 0), RES - 1);
  int y0c = min(max(y0, 0), RES - 1), y1c = min(max(y1, 0), RES - 1);
  uint32_t b00 = planeBase + (uint32_t)(y0c * RES + x0c) * (uint32_t)CH + (uint32_t)lane;
  uint32_t dx  = (uint32_t)(x1c - x0c) * (uint32_t)CH;
  uint32_t dy  = (uint32_t)(y1c - y0c) * (uint32_t)(RES * CH);
  float v00 = t[b00];
  float v01 = t[b00 + dx];
  float v10 = t[b00 + dy];
  float v11 = t[b00 + dy + dx];
  return fmaf(fmaf(v00, ax0, v01 * ax1), ay0, fmaf(v10, ax0, v11 * ax1) * ay1);
}

// Same math but on the original layout tp[((p*CH + c)*RES + y)*RES + x]
__device__ __forceinline__ float sample_plane_orig(const float* __restrict__ t,
                                                   uint32_t planeBase,
                                                   float gx, float gy, int lane) {
  float x = (gx + 1.0f) * (0.5f * (float)(RES - 1));
  float y = (gy + 1.0f) * (0.5f * (float)(RES - 1));
  float xf = floorf(x), yf = floorf(y);
  float wx1 = x - xf, wy1 = y - yf;
  float wx0 = 1.0f - wx1, wy0 = 1.0f - wy1;
  int x0 = (int)xf, y0 = (int)yf;
  int x1 = x0 + 1, y1 = y0 + 1;
  float ax0 = (x0 >= 0 && x0 < RES) ? wx0 : 0.0f;
  float ax1 = (x1 >= 0 && x1 < RES) ? wx1 : 0.0f;
  float ay0 = (y0 >= 0 && y0 < RES) ? wy0 : 0.0f;
  float ay1 = (y1 >= 0 && y1 < RES) ? wy1 : 0.0f;
  int x0c = min(max(x0, 0), RES - 1), x1c = min(max(x1, 0), RES - 1);
  int y0c = min(max(y0, 0), RES - 1), y1c = min(max(y1, 0), RES - 1);
  uint32_t b00 = planeBase + ((uint32_t)lane << 18) + (uint32_t)(y0c * RES + x0c);
  uint32_t dx  = (uint32_t)(x1c - x0c);
  uint32_t dy  = (uint32_t)(y1c - y0c) * (uint32_t)RES;
  float v00 = t[b00];
  float v01 = t[b00 + dx];
  float v10 = t[b00 + dy];
  float v11 = t[b00 + dy + dx];
  return fmaf(fmaf(v00, ax0, v01 * ax1), ay0, fmaf(v10, ax0, v11 * ax1) * ay1);
}

// ---------------------------------------------------------------------------
// Transpose [p][c][y][x] -> [p][y][x][c]  (channel-last) via LDS staging.
// One block per (p, y, 64-wide x-chunk). Phase 1 uses CDNA5 async
// global->LDS copies (ASYNCcnt); phase 2 reads LDS conflict-free and emits
// fully coalesced 128B channel-last stores.
// ---------------------------------------------------------------------------
__global__ __launch_bounds__(256) void triplane_transpose_kernel(
    const float* __restrict__ tp, const int* __restrict__ oid,
    float* __restrict__ wsT) {
  __shared__ float tile[64][33];  // [x][c], row stride 33 -> conflict-free both phases
  const int tid = threadIdx.x;
  const int blk = blockIdx.x;           // 0 .. 3*512*8 - 1
  const int xc  = blk & 7;
  const int y   = (blk >> 3) & (RES - 1);
  const int p   = blk >> 12;            // 0..2
  const int xb  = xc * 64;

  const float* src = tp + (size_t)oid[0] * (size_t)TRIPLANE_ELEMS;

  // Phase 1: 8 async copies per lane, one wait for all of them.
#pragma unroll
  for (int i = 0; i < 8; ++i) {
    int e = i * 256 + tid;              // 0..2047
    int x = e & 63;
    int c = e >> 6;                     // lanes of a wave share c -> coalesced 128B+
    uint32_t goff = ((uint32_t)p << 23) + ((uint32_t)c << 18) +
                    ((uint32_t)y << 9) + (uint32_t)(xb + x);
    uint64_t ga  = (uint64_t)(uintptr_t)(src + goff);
    uint32_t lds = (uint32_t)(uintptr_t)&tile[x][c];  // low 32 bits = LDS byte addr
    asm volatile("global_load_async_to_lds_b32 %0, %1, off"
                 :: "v"(lds), "v"(ga) : "memory");
  }
  asm volatile("s_wait_asynccnt 0" ::: "memory");
  __syncthreads();

  // Phase 2: lane = channel -> contiguous 128B stores; LDS reads stride-1.
#pragma unroll
  for (int i = 0; i < 8; ++i) {
    int e = i * 256 + tid;              // 0..2047
    int c = e & 31;
    int x = e >> 5;                     // fixed per wave
    uint32_t ooff = ((uint32_t)(((p * RES) + y) * RES + xb + x) << 5) + (uint32_t)c;
    wsT[ooff] = tile[x][c];
  }
}

// ---------------------------------------------------------------------------
// Gather: one wave per point, lane = channel. 12 coalesced 128B corner loads
// per point, NT output stores so the 96MB triplane stays resident in L2.
// ---------------------------------------------------------------------------
__global__ __launch_bounds__(256) void triplane_gather_cl(
    const float* __restrict__ xyz, const float* __restrict__ wsT,
    float* __restrict__ out, int M) {
  const int lane = threadIdx.x & 31;
  const int wave = blockIdx.x * (blockDim.x >> 5) + (threadIdx.x >> 5);
  const int nW   = gridDim.x * (blockDim.x >> 5);
  for (int m = wave; m < M; m += nW) {
    const float* pp = xyz + 3 * (size_t)m;
    float px = __builtin_nontemporal_load(pp + 0);
    float py = __builtin_nontemporal_load(pp + 1);
    float pz = __builtin_nontemporal_load(pp + 2);
    if (m + nW < M) __builtin_prefetch(xyz + 3 * (size_t)(m + nW), 0, 1);
    // plane0: (gx,gy)=(y,x); plane1: (z,x); plane2: (y,z)
    float acc;
    acc  = sample_plane_cl(wsT, 0u,                        py, px, lane);
    acc += sample_plane_cl(wsT, (uint32_t)PLANE_ELEMS,     pz, px, lane);
    acc += sample_plane_cl(wsT, (uint32_t)(2*PLANE_ELEMS), py, pz, lane);
    __builtin_nontemporal_store(acc, out + (size_t)m * CH + lane);
  }
}

// Fallback (workspace too small): correct but unoptimized, original layout.
__global__ __launch_bounds__(256) void triplane_gather_orig(
    const float* __restrict__ xyz, const float* __restrict__ tp,
    const int* __restrict__ oid, float* __restrict__ out, int M) {
  const float* t = tp + (size_t)oid[0] * (size_t)TRIPLANE_ELEMS;
  const int lane = threadIdx.x & 31;
  const int wave = blockIdx.x * (blockDim.x >> 5) + (threadIdx.x >> 5);
  const int nW   = gridDim.x * (blockDim.x >> 5);
  for (int m = wave; m < M; m += nW) {
    const float* pp = xyz + 3 * (size_t)m;
    float px = pp[0], py = pp[1], pz = pp[2];
    float acc;
    acc  = sample_plane_orig(t, 0u,                        py, px, lane);
    acc += sample_plane_orig(t, (uint32_t)PLANE_ELEMS,     pz, px, lane);
    acc += sample_plane_orig(t, (uint32_t)(2*PLANE_ELEMS), py, pz, lane);
    __builtin_nontemporal_store(acc, out + (size_t)m * CH + lane);
  }
}

extern "C" void kernel_launch(void* const* d_in, const int* in_sizes, int n_in,
                              void* d_out, int out_size, void* d_ws, size_t ws_size,
                              hipStream_t stream) {
  const float* xyz = (const float*)d_in[0];
  const float* tp  = (const float*)d_in[1];
  const int*   oid = (const int*)d_in[2];
  float* out = (float*)d_out;
  const int M = in_sizes[0] / 3;
  const size_t need = (size_t)TRIPLANE_ELEMS * sizeof(float);
  if (ws_size >= need) {
    triplane_transpose_kernel<<<3 * RES * (RES / 64), 256, 0, stream>>>(tp, oid, (float*)d_ws);
    triplane_gather_cl<<<4096, 256, 0, stream>>>(xyz, (const float*)d_ws, out, M);
  } else {
    triplane_gather_orig<<<4096, 256, 0, stream>>>(xyz, tp, oid, out, M);
  }
}